// SS3D_Conv_SSM_76252849373716
// MI455X (gfx1250) — compile-verified
//
#include <hip/hip_runtime.h>
#include <hip/hip_bf16.h>

// ---------------- problem constants ----------------
#define BB    2
#define DD    8
#define HH    16
#define WW    16
#define LSP   2048          // D*H*W
#define HID   128
#define HALF  64
#define DIN   128
#define NN    16
#define KK    6
#define RANK  4

typedef __attribute__((ext_vector_type(2))) float v2f;
typedef __attribute__((ext_vector_type(8))) float v8f;

static __device__ __forceinline__ v8f wmma4(v2f a, v2f b, v8f c) {
  // V_WMMA_F32_16X16X4_F32 : D = A(16x4) * B(4x16) + C(16x16), all fp32
  return __builtin_amdgcn_wmma_f32_16x16x4_f32(false, a, false, b, (short)0, c,
                                               false, false);
}

// CDNA5 async global->LDS copy (GV mode), tracked by ASYNCcnt.
// lds_off = byte offset into workgroup LDS (low 32 bits of generic ptr),
// gaddr   = 64-bit global address.
static __device__ __forceinline__ void async_g2l_b32(unsigned lds_off,
                                                     const float* gaddr) {
  asm volatile("global_load_async_to_lds_b32 %0, %1, off"
               :: "v"(lds_off), "v"(gaddr)
               : "memory");
}
static __device__ __forceinline__ void wait_async0() {
  asm volatile("s_wait_asynccnt 0x0" ::: "memory");
}

// map (direction k, scan position t) -> spatial index d*256+h*16+w
static __device__ __forceinline__ int spatial_of(int k, int t) {
  int s  = (k >= 3) ? (LSP - 1 - t) : t;
  int kk = (k >= 3) ? (k - 3) : k;
  if (kk == 0) return s;                    // (d,h,w) order
  int d, h, w;
  if (kk == 1) {                            // s = h*128 + w*8 + d
    d = s & 7; w = (s >> 3) & 15; h = s >> 7;
  } else {                                  // s = w*128 + d*16 + h
    h = s & 15; d = (s >> 4) & 7; w = s >> 7;
  }
  return d * 256 + h * 16 + w;
}

static __device__ __forceinline__ float warp_sum(float v) {
  for (int m = 16; m; m >>= 1) v += __shfl_xor(v, m, 32);
  return v;
}

// ---------------- fold BN into conv weights ----------------
__global__ void k_prep_w(const float* c1w, const float* c2w,
                         const float* bn1g, const float* bn1v,
                         const float* bn2g, const float* bn2v,
                         const float* bn3g, const float* bn3v,
                         float* w1f, float* w2f) {
  int co = blockIdx.x, ci = threadIdx.x;                 // 64 x 64
  float s1 = bn1g[ci] * rsqrtf(bn1v[ci] + 1e-5f);
  float s2 = bn2g[co] * rsqrtf(bn2v[co] + 1e-5f);
  float s3 = bn3g[co] * rsqrtf(bn3v[co] + 1e-5f);
  int base = (co * 64 + ci) * 27;
  for (int t = 0; t < 27; ++t) {
    w1f[base + t] = s2 * c1w[base + t] * s1;
    w2f[base + t] = s3 * c2w[base + t];
  }
}

__global__ void k_prep_b(const float* c1w, const float* c1b, const float* c2b,
                         const float* bn1g, const float* bn1b, const float* bn1m, const float* bn1v,
                         const float* bn2g, const float* bn2b, const float* bn2m, const float* bn2v,
                         const float* bn3g, const float* bn3b, const float* bn3m, const float* bn3v,
                         float* b1f, float* b2f) {
  int co = threadIdx.x;                                  // 64 threads, 1 block
  float s2 = bn2g[co] * rsqrtf(bn2v[co] + 1e-5f);
  float t2 = bn2b[co] - bn2m[co] * s2;
  float s3 = bn3g[co] * rsqrtf(bn3v[co] + 1e-5f);
  float t3 = bn3b[co] - bn3m[co] * s3;
  float acc = c1b[co];
  for (int ci = 0; ci < 64; ++ci) {
    float s1 = bn1g[ci] * rsqrtf(bn1v[ci] + 1e-5f);
    float t1 = bn1b[ci] - bn1m[ci] * s1;
    float ws = 0.f;
    int base = (co * 64 + ci) * 27;
    for (int t = 0; t < 27; ++t) ws += c1w[base + t];
    acc += ws * t1;
  }
  b1f[co] = s2 * acc + t2;
  b2f[co] = s3 * c2b[co] + t3;
}

// ---------------- LayerNorm of the "down" half ----------------
__global__ void k_ln_down(const float* inp, const float* g, const float* b,
                          float* out) {
  int wave = (blockIdx.x * blockDim.x + threadIdx.x) >> 5;   // one row per wave
  int lane = threadIdx.x & 31;
  const float* row = inp + (size_t)wave * HID + HALF;
  float v0 = row[lane], v1 = row[lane + 32];
  float mu = warp_sum(v0 + v1) * (1.f / 64.f);
  float d0 = v0 - mu, d1 = v1 - mu;
  float inv = rsqrtf(warp_sum(d0 * d0 + d1 * d1) * (1.f / 64.f) + 1e-6f);
  out[(size_t)wave * HALF + lane]      = d0 * inv * g[lane] + b[lane];
  out[(size_t)wave * HALF + lane + 32] = d1 * inv * g[lane + 32] + b[lane + 32];
}

// ---------------- in_proj: (4096x64) @ (64x256) via WMMA ----------------
__global__ void k_inproj(const float* X, const float* W, float* Y) {
  int wid  = (blockIdx.x * blockDim.x + threadIdx.x) >> 5;   // 256*16 waves
  int lane = threadIdx.x & 31;
  int mt = wid >> 4, nt = wid & 15;
  int m = lane & 15, g = lane >> 4;
  const float* arow = X + (size_t)(mt * 16 + m) * HALF;
  const float* bcol = W + (size_t)(nt * 16 + m) * HALF;      // W[n][k]
  v8f acc = {};
#pragma unroll
  for (int k0 = 0; k0 < HALF; k0 += 4) {
    v2f a, bb;
    int ka = k0 + 2 * g;
    a.x = arow[ka];  a.y = arow[ka + 1];
    bb.x = bcol[ka]; bb.y = bcol[ka + 1];
    acc = wmma4(a, bb, acc);
  }
#pragma unroll
  for (int r = 0; r < 8; ++r) {
    int row = mt * 16 + r + 8 * g;
    Y[(size_t)row * 256 + nt * 16 + (lane & 15)] = acc[r];
  }
}

// ---------------- depthwise 3^3 conv + SiLU -> xc (b,c,l) ----------------
__global__ void k_dwconv(const float* xz, const float* w, const float* bias,
                         float* xc) {
  int c  = threadIdx.x;                 // 128
  int sp = blockIdx.x;                  // B*L
  int b = sp >> 11, l = sp & 2047;
  int d = l >> 8, h = (l >> 4) & 15, wq = l & 15;
  float acc = bias[c];
  for (int dz = 0; dz < 3; ++dz) {
    int z = d + dz - 1; if ((unsigned)z >= (unsigned)DD) continue;
    for (int dy = 0; dy < 3; ++dy) {
      int y = h + dy - 1; if ((unsigned)y >= (unsigned)HH) continue;
      for (int dx = 0; dx < 3; ++dx) {
        int x = wq + dx - 1; if ((unsigned)x >= (unsigned)WW) continue;
        float v = xz[((size_t)(b * LSP + z * 256 + y * 16 + x)) * 256 + c];
        acc += v * w[c * 27 + dz * 9 + dy * 3 + dx];
      }
    }
  }
  float s = acc / (1.f + __expf(-acc));
  xc[((size_t)(b * DIN + c)) * LSP + l] = s;
}

// ---------------- x_dbl: per (b,k) (36x128)@(128x2048) via WMMA ----------
__global__ void k_xdbl(const float* xc, const float* xpw, float* xdbl) {
  int wid  = (blockIdx.x * blockDim.x + threadIdx.x) >> 5;   // 2*6*3*128
  int lane = threadIdx.x & 31;
  int nt = wid & 127; int tmp = wid >> 7;
  int mt = tmp % 3; tmp /= 3;
  int k = tmp % 6;  int b = tmp / 6;
  int m = lane & 15, g = lane >> 4;
  int dd = mt * 16 + m;
  int l  = nt * 16 + m;
  int sp = spatial_of(k, l);
  const float* xcb = xc + (size_t)b * DIN * LSP;
  const float* aw  = xpw + ((size_t)k * 36 + dd) * DIN;
  bool aval = dd < 36;
  v8f acc = {};
  for (int c0 = 0; c0 < DIN; c0 += 4) {
    int ca = c0 + 2 * g;
    v2f a, bb;
    a.x = aval ? aw[ca] : 0.f;
    a.y = aval ? aw[ca + 1] : 0.f;
    bb.x = xcb[(size_t)ca * LSP + sp];
    bb.y = xcb[(size_t)(ca + 1) * LSP + sp];
    acc = wmma4(a, bb, acc);
  }
  float* outb = xdbl + ((size_t)(b * 6 + k) * 36) * LSP;
#pragma unroll
  for (int r = 0; r < 8; ++r) {
    int ddr = mt * 16 + r + 8 * g;
    if (ddr < 36) outb[(size_t)ddr * LSP + nt * 16 + (lane & 15)] = acc[r];
  }
}

// ---------------- selective scan: block per (b,k), lane per channel -------
// (dt,B,C) coefficients for each 16-step chunk are staged into LDS with
// CDNA5 async global->LDS copies (ASYNCcnt), completed before each chunk's recurrence.
__global__ void k_scan(const float* xc, const float* xdbl, const float* dtw,
                       const float* dtb, const float* Alogs, const float* Ds,
                       float* ycomb) {
  int b = blockIdx.x / 6, k = blockIdx.x % 6;
  int c = threadIdx.x;                              // 128
  float A[NN], h[NN];
#pragma unroll
  for (int n = 0; n < NN; ++n) {
    A[n] = -__expf(Alogs[(size_t)(k * DIN + c) * NN + n]);
    h[n] = 0.f;
  }
  float w0 = dtw[(k * DIN + c) * 4 + 0], w1 = dtw[(k * DIN + c) * 4 + 1];
  float w2 = dtw[(k * DIN + c) * 4 + 2], w3 = dtw[(k * DIN + c) * 4 + 3];
  float dtbv = dtb[k * DIN + c];
  float Dv   = Ds[k * DIN + c];
  const float* xd  = xdbl + (size_t)(b * 6 + k) * 36 * LSP;
  const float* xcc = xc + ((size_t)(b * DIN + c)) * LSP;
  float* yb = ycomb + (size_t)b * LSP * DIN;

  __shared__ float sd[36 * 16];                     // (dt,B,C) for 16 steps
  for (int t0 = 0; t0 < LSP; t0 += 16) {
    __syncthreads();                                // prior chunk fully read
    for (int idx = c; idx < 36 * 16; idx += DIN) {
      int ddi = idx >> 4, tt = idx & 15;
      unsigned lds_off = (unsigned)(size_t)(&sd[idx]);
      async_g2l_b32(lds_off, &xd[(size_t)ddi * LSP + t0 + tt]);
    }
    if (c < 36 && t0 + 16 < LSP)                    // prefetch next chunk
      __builtin_prefetch(&xd[(size_t)c * LSP + t0 + 16], 0, 0);
    wait_async0();                                  // my wave's copies done
    __syncthreads();                                // all waves' copies done
    for (int tt = 0; tt < 16; ++tt) {
      int t = t0 + tt;
      int sp = spatial_of(k, t);
      float u = xcc[sp];
      float dr = w0 * sd[tt] + w1 * sd[16 + tt] + w2 * sd[32 + tt] +
                 w3 * sd[48 + tt] + dtbv;
      float delta = (dr > 20.f) ? dr : __logf(1.f + __expf(dr));
      float du = delta * u;
      float y = Dv * u;
#pragma unroll
      for (int n = 0; n < NN; ++n) {
        float Bn = sd[(4 + n) * 16 + tt];
        float Cn = sd[(20 + n) * 16 + tt];
        h[n] = __expf(delta * A[n]) * h[n] + du * Bn;
        y += h[n] * Cn;
      }
      atomicAdd(&yb[(size_t)sp * DIN + c], y);      // merge 6 directions
    }
  }
}

// ---------------- out LayerNorm + SiLU gate (in place) -------------------
__global__ void k_lngate(float* ycomb, const float* g, const float* bb,
                         const float* xz) {
  int wave = (blockIdx.x * blockDim.x + threadIdx.x) >> 5;   // row per wave
  int lane = threadIdx.x & 31;
  float* row = ycomb + (size_t)wave * DIN;
  float v[4]; float s = 0.f;
#pragma unroll
  for (int i = 0; i < 4; ++i) { v[i] = row[lane + 32 * i]; s += v[i]; }
  float mu = warp_sum(s) * (1.f / 128.f);
  float q = 0.f;
#pragma unroll
  for (int i = 0; i < 4; ++i) { v[i] -= mu; q += v[i] * v[i]; }
  float inv = rsqrtf(warp_sum(q) * (1.f / 128.f) + 1e-5f);
  const float* zrow = xz + (size_t)wave * 256 + DIN;
#pragma unroll
  for (int i = 0; i < 4; ++i) {
    int ch = lane + 32 * i;
    float z = zrow[ch];
    float sil = z / (1.f + __expf(-z));
    row[ch] = (v[i] * inv * g[ch] + bb[ch]) * sil;
  }
}

// ---------------- out_proj (4096x128)@(128x64) -> odd channels + residual
__global__ void k_outproj(const float* Yg, const float* W, const float* inp,
                          float* out) {
  int wid  = (blockIdx.x * blockDim.x + threadIdx.x) >> 5;   // 256*4 waves
  int lane = threadIdx.x & 31;
  int mt = wid >> 2, nt = wid & 3;
  int m = lane & 15, g = lane >> 4;
  const float* arow = Yg + (size_t)(mt * 16 + m) * DIN;
  const float* bcol = W + (size_t)(nt * 16 + m) * DIN;
  v8f acc = {};
#pragma unroll 8
  for (int k0 = 0; k0 < DIN; k0 += 4) {
    int ka = k0 + 2 * g;
    v2f a, bb;
    a.x = arow[ka];  a.y = arow[ka + 1];
    bb.x = bcol[ka]; bb.y = bcol[ka + 1];
    acc = wmma4(a, bb, acc);
  }
  int col = nt * 16 + (lane & 15);
#pragma unroll
  for (int r = 0; r < 8; ++r) {
    int row = mt * 16 + r + 8 * g;
    size_t o = (size_t)row * HID + 2 * col + 1;
    out[o] = acc[r] + inp[o];
  }
}

// ---------------- 3^3 conv as implicit GEMM (WMMA), fused BN + ReLU ------
__global__ void k_conv3(const float* in, int instride, const float* wf,
                        const float* bf, float* out) {
  int wid  = (blockIdx.x * blockDim.x + threadIdx.x) >> 5;   // 256*4 waves
  int lane = threadIdx.x & 31;
  int nt = wid & 3; int mth = wid >> 2;                      // (b,d,h) tile
  int b = mth >> 7, d = (mth >> 4) & 7, h = mth & 15;
  int m = lane & 15, g = lane >> 4;                          // m = w coord
  int n = nt * 16 + m;                                       // out channel
  const float* inb = in + (size_t)b * LSP * instride;
  v8f acc = {};
  for (int dz = 0; dz < 3; ++dz) {
    int z = d + dz - 1; if ((unsigned)z >= (unsigned)DD) continue;
    for (int dy = 0; dy < 3; ++dy) {
      int y = h + dy - 1; if ((unsigned)y >= (unsigned)HH) continue;
      for (int dx = 0; dx < 3; ++dx) {
        int x = m + dx - 1;
        bool ok = (unsigned)x < (unsigned)WW;
        const float* src = inb + (size_t)(z * 256 + y * 16 + x) * instride;
        int tap = dz * 9 + dy * 3 + dx;
        const float* wcol = wf + (size_t)n * 64 * 27 + tap;  // + ci*27
#pragma unroll
        for (int c0 = 0; c0 < 64; c0 += 4) {
          int ca = c0 + 2 * g;
          v2f a, bb;
          a.x = ok ? src[ca] : 0.f;
          a.y = ok ? src[ca + 1] : 0.f;
          bb.x = wcol[(size_t)ca * 27];
          bb.y = wcol[(size_t)(ca + 1) * 27];
          acc = wmma4(a, bb, acc);
        }
      }
    }
  }
  float bias = bf[n];
  float* outr = out + ((size_t)(b * LSP + d * 256 + h * 16)) * 64;
#pragma unroll
  for (int r = 0; r < 8; ++r) {
    int wrow = r + 8 * g;
    float v = acc[r] + bias;
    outr[(size_t)wrow * 64 + nt * 16 + (lane & 15)] = v > 0.f ? v : 0.f;
  }
}

// ---------------- 1x1 conv + ReLU -> even channels + residual ------------
__global__ void k_conv1x1(const float* u2, const float* W, const float* bias,
                          const float* inp, float* out) {
  int wid  = (blockIdx.x * blockDim.x + threadIdx.x) >> 5;   // 256*4 waves
  int lane = threadIdx.x & 31;
  int mt = wid >> 2, nt = wid & 3;
  int m = lane & 15, g = lane >> 4;
  const float* arow = u2 + (size_t)(mt * 16 + m) * 64;
  const float* bcol = W + (size_t)(nt * 16 + m) * 64;        // c3_w[n][ci]
  v8f acc = {};
#pragma unroll
  for (int k0 = 0; k0 < 64; k0 += 4) {
    int ka = k0 + 2 * g;
    v2f a, bb;
    a.x = arow[ka];  a.y = arow[ka + 1];
    bb.x = bcol[ka]; bb.y = bcol[ka + 1];
    acc = wmma4(a, bb, acc);
  }
  int col = nt * 16 + (lane & 15);
  float bv = bias[col];
#pragma unroll
  for (int r = 0; r < 8; ++r) {
    int row = mt * 16 + r + 8 * g;
    float v = acc[r] + bv;
    v = v > 0.f ? v : 0.f;
    size_t o = (size_t)row * HID + 2 * col;
    out[o] = v + inp[o];
  }
}

// ---------------- host launcher ----------------
extern "C" void kernel_launch(void* const* d_in, const int* in_sizes, int n_in,
                              void* d_out, int out_size, void* d_ws,
                              size_t ws_size, hipStream_t stream) {
  const float* input     = (const float*)d_in[0];
  const float* ln1_g     = (const float*)d_in[1];
  const float* ln1_b     = (const float*)d_in[2];
  const float* in_proj_w = (const float*)d_in[3];
  const float* conv_w    = (const float*)d_in[4];
  const float* conv_b    = (const float*)d_in[5];
  const float* x_proj_w  = (const float*)d_in[6];
  const float* dt_w      = (const float*)d_in[7];
  const float* dt_b      = (const float*)d_in[8];
  const float* A_logs    = (const float*)d_in[9];
  const float* Ds        = (const float*)d_in[10];
  const float* out_ng    = (const float*)d_in[11];
  const float* out_nb    = (const float*)d_in[12];
  const float* out_pw    = (const float*)d_in[13];
  const float* bn1_g = (const float*)d_in[14], *bn1_b = (const float*)d_in[15];
  const float* bn1_m = (const float*)d_in[16], *bn1_v = (const float*)d_in[17];
  const float* bn2_g = (const float*)d_in[18], *bn2_b = (const float*)d_in[19];
  const float* bn2_m = (const float*)d_in[20], *bn2_v = (const float*)d_in[21];
  const float* bn3_g = (const float*)d_in[22], *bn3_b = (const float*)d_in[23];
  const float* bn3_m = (const float*)d_in[24], *bn3_v = (const float*)d_in[25];
  const float* c1_w = (const float*)d_in[26], *c1_b = (const float*)d_in[27];
  const float* c2_w = (const float*)d_in[28], *c2_b = (const float*)d_in[29];
  const float* c3_w = (const float*)d_in[30], *c3_b = (const float*)d_in[31];
  float* out = (float*)d_out;

  float* ws = (float*)d_ws;
  float* down_ln = ws;                         // 2*2048*64
  float* xz      = down_ln + 262144;           // 2*2048*256
  float* xc      = xz + 1048576;               // 2*128*2048
  float* xdbl    = xc + 524288;                // 2*6*36*2048
  float* ycomb   = xdbl + 884736;              // 2*2048*128
  float* u1      = ycomb + 524288;             // 2*2048*64
  float* u2      = u1 + 262144;                // 2*2048*64
  float* w1f     = u2 + 262144;                // 64*64*27
  float* b1f     = w1f + 110592;               // 64
  float* w2f     = b1f + 64;                   // 64*64*27
  float* b2f     = w2f + 110592;               // 64

  hipMemsetAsync(ycomb, 0, (size_t)524288 * sizeof(float), stream);

  k_prep_w<<<64, 64, 0, stream>>>(c1_w, c2_w, bn1_g, bn1_v, bn2_g, bn2_v,
                                  bn3_g, bn3_v, w1f, w2f);
  k_prep_b<<<1, 64, 0, stream>>>(c1_w, c1_b, c2_b, bn1_g, bn1_b, bn1_m, bn1_v,
                                 bn2_g, bn2_b, bn2_m, bn2_v, bn3_g, bn3_b,
                                 bn3_m, bn3_v, b1f, b2f);

  // SSM branch
  k_ln_down<<<512, 256, 0, stream>>>(input, ln1_g, ln1_b, down_ln);
  k_inproj<<<512, 256, 0, stream>>>(down_ln, in_proj_w, xz);   // 4096 waves
  k_dwconv<<<BB * LSP, DIN, 0, stream>>>(xz, conv_w, conv_b, xc);
  k_xdbl<<<576, 256, 0, stream>>>(xc, x_proj_w, xdbl);          // 4608 waves
  k_scan<<<BB * KK, DIN, 0, stream>>>(xc, xdbl, dt_w, dt_b, A_logs, Ds, ycomb);
  k_lngate<<<512, 256, 0, stream>>>(ycomb, out_ng, out_nb, xz);
  k_outproj<<<128, 256, 0, stream>>>(ycomb, out_pw, input, out); // 1024 waves

  // Conv branch (BN folded)
  k_conv3<<<128, 256, 0, stream>>>(input, HID, w1f, b1f, u1);    // conv1
  k_conv3<<<128, 256, 0, stream>>>(u1, 64, w2f, b2f, u2);        // conv2
  k_conv1x1<<<128, 256, 0, stream>>>(u2, c3_w, c3_b, input, out);
}